// ThimbleLayer_12017318494900
// MI455X (gfx1250) — compile-verified
//
#include <hip/hip_runtime.h>
#include <math.h>

// ---------------- problem constants ----------------
#define LDIM   24
#define L2S    (LDIM*LDIM)          // 576
#define L3S    (LDIM*LDIM*LDIM)     // 13824
#define L4S    (LDIM*LDIM*LDIM*LDIM)// 331776
#define NB     4
#define NMCS   16
#define NSAMP  64                   // NB*NMCS
#define CH     16                   // chunks (blocks) per sample
#define SITES_PER_CHUNK (L4S / CH)  // 20736
#define TPB    256
#define ITERS  (SITES_PER_CHUNK / TPB) // 81
#define NQ     13                   // partial quantities per sample
#define PSTRIDE 16                  // padded partial stride

typedef float v2f __attribute__((ext_vector_type(2)));
typedef float v8f __attribute__((ext_vector_type(8)));

// ---------------- complex helpers ----------------
struct Cpx { float re, im; };
__device__ __forceinline__ Cpx cadd(Cpx a, Cpx b){ return {a.re+b.re, a.im+b.im}; }
__device__ __forceinline__ Cpx csub(Cpx a, Cpx b){ return {a.re-b.re, a.im-b.im}; }
__device__ __forceinline__ Cpx cmul(Cpx a, Cpx b){ return {fmaf(a.re,b.re,-a.im*b.im), fmaf(a.re,b.im, a.im*b.re)}; }
__device__ __forceinline__ Cpx csq (Cpx a){ return {fmaf(a.re,a.re,-a.im*a.im), 2.f*a.re*a.im}; }
__device__ __forceinline__ Cpx cscale(float s, Cpx a){ return {s*a.re, s*a.im}; }

// =====================================================================
// Kernel 1: fused per-site pass (action + logdet + observables),
//           block-level LDS tree reduction -> 13 partials per block.
// =====================================================================
__global__ __launch_bounds__(TPB) void k_partials(
    const float* __restrict__ phi_re, const float* __restrict__ phi_im,
    const float* __restrict__ ker,    const float* __restrict__ lag,
    float* __restrict__ part)
{
    const int blk   = blockIdx.x;
    const int n     = blk / CH;        // sample index
    const int chunk = blk % CH;
    const int tid   = threadIdx.x;

    const float* __restrict__ pr = phi_re + (size_t)n * L4S;
    const float* __restrict__ pi = phi_im + (size_t)n * L4S;
    const float* __restrict__ kv = ker    + (size_t)n * L4S;
    const float* __restrict__ P  = lag + (n / NMCS) * 8;

    const float m2 = P[0], kin = P[1], g3 = P[2], g4 = P[3];
    const float g6 = P[4], boxc = P[5], dint = P[6], mu = P[7];
    const float emu = __expf(mu), emn = __expf(-mu);
    const float c3 = g3 * (1.f/6.f), c4 = g4 * (1.f/24.f), c6 = g6 * (1.f/720.f);

    float acc[NQ];
#pragma unroll
    for (int q = 0; q < NQ; ++q) acc[q] = 0.f;

    const int base = chunk * SITES_PER_CHUNK + tid;
    for (int it = 0; it < ITERS; ++it) {
        const int s  = base + it * TPB;
        const int z  = s % LDIM;
        const int sy = s / LDIM;
        const int y  = sy % LDIM;
        const int sx = sy / LDIM;
        const int x  = sx % LDIM;
        const int t  = sx / LDIM;

        // periodic neighbor linear indices
        const int tp = (t == LDIM-1) ? s - (LDIM-1)*L3S : s + L3S;
        const int tm = (t == 0)      ? s + (LDIM-1)*L3S : s - L3S;
        const int xp = (x == LDIM-1) ? s - (LDIM-1)*L2S : s + L2S;
        const int xm = (x == 0)      ? s + (LDIM-1)*L2S : s - L2S;
        const int yp = (y == LDIM-1) ? s - (LDIM-1)*LDIM : s + LDIM;
        const int ym = (y == 0)      ? s + (LDIM-1)*LDIM : s - LDIM;
        const int zp = (z == LDIM-1) ? s - (LDIM-1)      : s + 1;
        const int zm = (z == 0)      ? s + (LDIM-1)      : s - 1;

        const Cpx c   = { pr[s],  pi[s]  };
        const Cpx atp = { pr[tp], pi[tp] };
        const Cpx atm = { pr[tm], pi[tm] };
        const Cpx axp = { pr[xp], pi[xp] };
        const Cpx axm = { pr[xm], pi[xm] };
        const Cpx ayp = { pr[yp], pi[yp] };
        const Cpx aym = { pr[ym], pi[ym] };
        const Cpx azp = { pr[zp], pi[zp] };
        const Cpx azm = { pr[zm], pi[zm] };
        const float kval = kv[s];

        // ---- action ----
        const Cpx dt  = csub(atp, c);
        const Cpx dxv = csub(axp, c);
        const Cpx dyv = csub(ayp, c);
        const Cpx dzv = csub(azp, c);
        const Cpx g2  = cadd(cadd(csq(dt), csq(dxv)), cadd(csq(dyv), csq(dzv)));
        const Cpx p2  = csq(c);
        const Cpx p3  = cmul(p2, c);
        const Cpx p4  = csq(p2);
        const Cpx p6  = cmul(p4, p2);

        Cpx S = cscale(0.5f * kin, g2);
        S = cadd(S, cscale(0.5f * m2, p2));
        S = cadd(S, cscale(c3, p3));
        S = cadd(S, cscale(c4, p4));
        S = cadd(S, cscale(c6, p6));

        Cpx box = cadd(cadd(cadd(atp, atm), cadd(axp, axm)),
                       cadd(cadd(ayp, aym), cadd(azp, azm)));
        box = csub(box, cscale(8.f, c));
        S = cadd(S, cscale(boxc, csq(box)));
        S = cadd(S, cscale(dint, cmul(p2, g2)));

        const Cpx hop = cadd(cscale(emu, cmul(atp, c)), cscale(emn, cmul(atm, c)));
        S = csub(S, cscale(0.5f, hop));

        acc[0] += S.re;
        acc[1] += S.im;
        acc[2] += -0.5f * __logf(kval);

        // ---- per-site observables ----
        const Cpx ps = { fminf(fmaxf(c.re, -20.f), 20.f),
                         fminf(fmaxf(c.im, -20.f), 20.f) };
        const Cpx q2 = csq(ps);
        const Cpx q4 = csq(q2);
        acc[3] += ps.re;  acc[4] += ps.im;
        acc[5] += q2.re;  acc[6] += q2.im;
        acc[7] += q4.re;  acc[8] += q4.im;

        const Cpx gz = csub(azp, c);         // roll(-1, axis=-1)
        const Cpx gy = csub(ayp, c);         // roll(-1, axis=-2)
        const Cpx gs = cadd(csq(gz), csq(gy));
        acc[9]  += gs.re; acc[10] += gs.im;

        const Cpx corr = cmul(c, azm);       // phi * roll(phi, +1, axis=-1)
        acc[11] += corr.re; acc[12] += corr.im;
    }

    // ---- block reduction (fixed order -> deterministic) ----
    __shared__ float red[NQ][TPB];
#pragma unroll
    for (int q = 0; q < NQ; ++q) red[q][tid] = acc[q];
    __syncthreads();
    for (int off = TPB / 2; off > 0; off >>= 1) {
        if (tid < off) {
#pragma unroll
            for (int q = 0; q < NQ; ++q) red[q][tid] += red[q][tid + off];
        }
        __syncthreads();
    }
    if (tid == 0) {
        float* o = part + (size_t)blk * PSTRIDE;
#pragma unroll
        for (int q = 0; q < NQ; ++q) o[q] = red[q][0];
    }
}

// =====================================================================
// Kernel 2: per-sample finalize + group softmax -> w[64], obs[64][16]
// =====================================================================
__global__ __launch_bounds__(NSAMP) void k_finalize(
    const float* __restrict__ part, const float* __restrict__ log_det_im,
    float* __restrict__ wout, float* __restrict__ obsmat)
{
    const int j = threadIdx.x;   // sample 0..63
    float acc[NQ];
#pragma unroll
    for (int q = 0; q < NQ; ++q) acc[q] = 0.f;
    for (int cchunk = 0; cchunk < CH; ++cchunk) {
        const float* p = part + (size_t)(j * CH + cchunk) * PSTRIDE;
#pragma unroll
        for (int q = 0; q < NQ; ++q) acc[q] += p[q];
    }

    const float S_im = acc[1];
    const float logw = -acc[0] + acc[2] + log_det_im[j];
    const float inv  = 1.f / (float)L4S;

    float obs[12];
    obs[0] = acc[3] * inv;  obs[1] = acc[4] * inv;
    obs[2] = acc[5] * inv;  obs[3] = acc[6] * inv;
    obs[4] = acc[7] * inv;  obs[5] = acc[8] * inv;
    obs[6] = cosf(S_im);    obs[7] = sinf(S_im);
    obs[8]  = fminf(fmaxf(acc[9]  * inv, -50.f), 50.f);
    obs[9]  = fminf(fmaxf(acc[10] * inv, -50.f), 50.f);
    obs[10] = fminf(fmaxf(acc[11] * inv, -20.f), 20.f);
    obs[11] = fminf(fmaxf(acc[12] * inv, -20.f), 20.f);

    // stable softmax over the 16 MC samples of this batch
    __shared__ float lw[NSAMP];
    __shared__ float ee[NSAMP];
    lw[j] = logw;
    __syncthreads();
    const int b0 = (j / NMCS) * NMCS;
    float m = -3.4e38f;
    for (int k = 0; k < NMCS; ++k) m = fmaxf(m, lw[b0 + k]);
    const float e = __expf(logw - m);
    ee[j] = e;
    __syncthreads();
    float ssum = 0.f;
    for (int k = 0; k < NMCS; ++k) ssum += ee[b0 + k];
    wout[j] = e / ssum;

    float* row = obsmat + j * 16;
#pragma unroll
    for (int f = 0; f < 12; ++f) row[f] = obs[f];
    row[12] = 0.f; row[13] = 0.f; row[14] = 0.f; row[15] = 0.f;
}

// =====================================================================
// Kernel 3: single wave32 -> feats[b,f] = sum_j w[b,j]*obs[b,j,f]
// Exact f32 weighted sum on the matrix pipe: per batch, 4 accumulating
// V_WMMA_F32_16X16X4_F32 (K=4 chunks over the 16 MC samples).
//   A[m,k] = w chunk (rows replicated), B[k,n] = obs rows, D = feats.
// A/B half-wave layout: lanes 16-31 hold K += 2 (per ISA 16x4 f32 table).
// =====================================================================
__global__ __launch_bounds__(32) void k_wsum(
    const float* __restrict__ w, const float* __restrict__ obsmat,
    float* __restrict__ out)
{
    const int l = threadIdx.x;
#if __has_builtin(__builtin_amdgcn_wmma_f32_16x16x4_f32)
    const int hi = (l >= 16) ? 1 : 0;
    const int n  = l & 15;
    for (int b = 0; b < NB; ++b) {
        v8f c = {0.f,0.f,0.f,0.f,0.f,0.f,0.f,0.f};
#pragma unroll
        for (int kk = 0; kk < 4; ++kk) {
            const float* w4 = w + b * NMCS + kk * 4;
            v2f a, bm;
            a.x  = w4[hi ? 2 : 0];
            a.y  = w4[hi ? 3 : 1];
            bm.x = obsmat[(b * NMCS + kk * 4 + (hi ? 2 : 0)) * 16 + n];
            bm.y = obsmat[(b * NMCS + kk * 4 + (hi ? 3 : 1)) * 16 + n];
            c = __builtin_amdgcn_wmma_f32_16x16x4_f32(
                    /*neg_a=*/false, a, /*neg_b=*/false, bm,
                    /*c_mod=*/(short)0, c, /*reuse_a=*/false, /*reuse_b=*/false);
        }
        if (l < 12) out[b * 12 + l] = c[0];   // row 0 of D = feats[b, :]
    }
#else
    if (l < 12) {
        for (int b = 0; b < NB; ++b) {
            float sum = 0.f;
            for (int j = 0; j < NMCS; ++j)
                sum += w[b * NMCS + j] * obsmat[(b * NMCS + j) * 16 + l];
            out[b * 12 + l] = sum;
        }
    }
#endif
}

// =====================================================================
extern "C" void kernel_launch(void* const* d_in, const int* in_sizes, int n_in,
                              void* d_out, int out_size, void* d_ws, size_t ws_size,
                              hipStream_t stream)
{
    const float* phi_free   = (const float*)d_in[0];
    const float* phi_im     = (const float*)d_in[1];
    const float* kernel_in  = (const float*)d_in[2];
    const float* log_det_im = (const float*)d_in[3];
    const float* lag        = (const float*)d_in[4];
    float* out = (float*)d_out;

    float* W    = (float*)d_ws;
    float* part = W;                                   // NSAMP*CH*PSTRIDE floats
    float* wbuf = W + (size_t)NSAMP * CH * PSTRIDE;    // NSAMP floats
    float* obsm = wbuf + NSAMP;                        // NSAMP*16 floats

    k_partials<<<NSAMP * CH, TPB, 0, stream>>>(phi_free, phi_im, kernel_in, lag, part);
    k_finalize<<<1, NSAMP, 0, stream>>>(part, log_det_im, wbuf, obsm);
    k_wsum<<<1, 32, 0, stream>>>(wbuf, obsm, out);
}